// gnn_last_44538810860107
// MI455X (gfx1250) — compile-verified
//
#include <hip/hip_runtime.h>

typedef __attribute__((ext_vector_type(2))) float v2f;
typedef __attribute__((ext_vector_type(8))) float v8f;

#define B_      8
#define N_      2048
#define J_      4
#define F_      16
#define C_      2
#define MCHUNKS 16
#define MC_LEN  (N_ / MCHUNKS)          // 128 m-steps per chunk
#define NT32    (N_ / 32)               // 64 n-tiles of width 32
#define OUT_ELEMS (B_ * N_ * C_)        // 32768

// ---------------------------------------------------------------------------
// Kernel 1: fold conv_w into x.
//   u2[b][m][c*4+j] = sum_f conv_w[c, j*16+f] * x[b,f,m]
// 131072 threads, trivial cost.
// ---------------------------------------------------------------------------
__global__ __launch_bounds__(256) void k_prep(const float* __restrict__ x,
                                              const float* __restrict__ cw,
                                              float* __restrict__ u2) {
  unsigned tid = blockIdx.x * 256u + threadIdx.x;       // [0, 131072)
  unsigned q = tid & 7u;                                // c*4 + j
  unsigned m = (tid >> 3) & (N_ - 1u);
  unsigned b = tid >> 14;
  unsigned c = q >> 2, j = q & 3u;
  const float* xp = x + ((size_t)b * F_) * N_ + m;      // x[b, f, m], f-stride N_
  const float* wp = cw + c * (F_ * J_) + j * F_;        // conv_w[c, j*16 + f]
  float s = 0.f;
#pragma unroll
  for (int f = 0; f < F_; ++f)
    s = __builtin_fmaf(wp[f], xp[(size_t)f * N_], s);
  u2[tid] = s;
}

// ---------------------------------------------------------------------------
// Kernel 2: main W stream with V_WMMA_F32_16X16X4_F32.
// One wave handles (b, 32-wide n-tile, m-chunk of 128):
//   per m: A = u2[b][m][*] (K=j, M=c), B0/B1 = W[b][m][n0..n0+31][0..3]
//   D(16x16) accumulates out[c, n_local] over the chunk.
// Each B load: 32 lanes x b64 == one fully-coalesced 256B request.
// Partials stored to ws (deterministic; reduced in kernel 3).
// ---------------------------------------------------------------------------
__global__ __launch_bounds__(256) void k_main(const float* __restrict__ Wt,
                                              const float* __restrict__ u2,
                                              float* __restrict__ partial) {
  const unsigned lane = threadIdx.x & 31u;
  const unsigned wave = blockIdx.x * (blockDim.x >> 5) + (threadIdx.x >> 5); // [0, 8192)

  const unsigned mc = wave & (MCHUNKS - 1u);        // m-chunk   [0,16)
  const unsigned nt = (wave >> 4) & (NT32 - 1u);    // n-tile    [0,64)
  const unsigned b  = wave >> 10;                   // batch     [0,8)
  const unsigned n0 = nt * 32u;
  const unsigned m0 = mc * MC_LEN;

  const unsigned lmod = lane & 15u;
  const unsigned lhi  = lane >> 4;

  // B-matrix source: &W[b][m][n0 + lmod][2*lhi]  (b64 per lane, 256B/wave)
  const float* pB = Wt + (((size_t)b * N_ + m0) * N_ + n0) * J_ + lmod * J_ + lhi * 2u;
  // A-matrix source: u2[b][m][cidx*4 + 2*lhi] ; rows c>=2 are don't-care (clamped addr)
  const unsigned aoff = ((lmod < 2u) ? lmod * 4u : 0u) + lhi * 2u;
  const float* pu = u2 + ((size_t)b * N_ + m0) * 8u + aoff;

  v8f d0 = {}, d1 = {};
#pragma unroll 4
  for (int m = 0; m < MC_LEN; ++m) {
    v2f a  = *(const v2f*)pu;
    v2f b0 = __builtin_nontemporal_load((const v2f*)pB);        // W read exactly once
    v2f b1 = __builtin_nontemporal_load((const v2f*)(pB + 64)); // n0+16 tile
    d0 = __builtin_amdgcn_wmma_f32_16x16x4_f32(false, a, false, b0, (short)0, d0,
                                               false, false);
    d1 = __builtin_amdgcn_wmma_f32_16x16x4_f32(false, a, false, b1, (short)0, d1,
                                               false, false);
    pB += (size_t)N_ * J_;   // next m row: +32 KiB
    pu += 8;
  }

  // D layout: VGPR0 lanes0-15 = row c=0 (N=lane), VGPR1 lanes0-15 = row c=1.
  if (lane < 16u) {
    float* po = partial + (size_t)mc * OUT_ELEMS + ((size_t)b * N_ + n0 + lane) * C_;
    v2f r0; r0.x = d0[0]; r0.y = d0[1];
    v2f r1; r1.x = d1[0]; r1.y = d1[1];
    *(v2f*)po         = r0;                 // n = n0 + lane
    *(v2f*)(po + 32)  = r1;                 // n = n0 + 16 + lane
  }
}

// ---------------------------------------------------------------------------
// Kernel 3: deterministic reduction of the 16 m-chunk partials + bias.
// ---------------------------------------------------------------------------
__global__ __launch_bounds__(256) void k_finish(const float* __restrict__ partial,
                                                const float* __restrict__ cb,
                                                float* __restrict__ out) {
  unsigned tid = blockIdx.x * 256u + threadIdx.x;   // [0, 32768)
  float s = cb[tid & 1u];
#pragma unroll
  for (int mc = 0; mc < MCHUNKS; ++mc)
    s += partial[(size_t)mc * OUT_ELEMS + tid];
  out[tid] = s;
}

// ---------------------------------------------------------------------------
extern "C" void kernel_launch(void* const* d_in, const int* in_sizes, int n_in,
                              void* d_out, int out_size, void* d_ws, size_t ws_size,
                              hipStream_t stream) {
  const float* W  = (const float*)d_in[0];   // [B, N, N, J] fp32, 512 MiB
  const float* x  = (const float*)d_in[1];   // [B, F, N, 1] fp32
  const float* cw = (const float*)d_in[2];   // [C, F*J]     fp32
  const float* cb = (const float*)d_in[3];   // [C]          fp32
  float* out = (float*)d_out;                // [B, N, C]    fp32

  float* u2      = (float*)d_ws;             // 8*2048*8   floats = 512 KiB
  float* partial = u2 + (size_t)B_ * N_ * 8; // 16*32768   floats = 2 MiB

  // 1) fold conv into x
  k_prep<<<(B_ * N_ * 8) / 256, 256, 0, stream>>>(x, cw, u2);
  // 2) stream W through WMMA: 8192 waves = 1024 blocks x 8 waves
  k_main<<<(B_ * NT32 * MCHUNKS * 32) / 256, 256, 0, stream>>>(W, u2, partial);
  // 3) reduce partials + bias
  k_finish<<<OUT_ELEMS / 256, 256, 0, stream>>>(partial, cb, out);
}